// DCNv2_91268055040059
// MI455X (gfx1250) — compile-verified
//
#include <hip/hip_runtime.h>
#include <math.h>

typedef __attribute__((ext_vector_type(2))) float v2f;
typedef __attribute__((ext_vector_type(8))) float v8f;

#define Bn   8
#define CIN  64
#define COUT 64
#define Hs   128
#define Ws   128
#define HW   (Hs * Ws)      // 16384
#define NPOS (Bn * HW)      // 131072
#define KTOT 576            // 9 taps * 64 channels

// ---------------------------------------------------------------------------
// Kernel 0: NCHW -> NHWC transpose of x (channel rows become contiguous 256B
// so bilinear gathers / A-fragment loads are b64-vectorizable).
// ---------------------------------------------------------------------------
__global__ __launch_bounds__(256) void nchw_to_nhwc(const float* __restrict__ x,
                                                    float* __restrict__ xt) {
    int idx = blockIdx.x * 256 + threadIdx.x;   // ((b*H+h)*W+w)*64 + c
    int c  = idx & 63;
    int p  = idx >> 6;
    int b  = p >> 14;
    int hw = p & (HW - 1);
    xt[idx] = x[((size_t)(b * CIN + c)) * HW + hw];
}

// ---------------------------------------------------------------------------
// Kernel 1: offset+mask conv weights -> pair-interleaved K-major B panel.
// Layout: wcomb[((K>>1)*32 + oc)*2 + (K&1)]  with K = tap*64 + c.
// A lane's B fragment (rows K+2kh, K+2kh+1, col n) is then ONE b64 load.
// ---------------------------------------------------------------------------
__global__ __launch_bounds__(256) void pack_conv_w(const float* __restrict__ ow,
                                                   const float* __restrict__ ob,
                                                   const float* __restrict__ mw,
                                                   const float* __restrict__ mb,
                                                   float* __restrict__ wcomb,
                                                   float* __restrict__ bias32) {
    int idx = blockIdx.x * 256 + threadIdx.x;
    if (idx >= KTOT * 32) return;
    int oc  = idx & 31;
    int k   = idx >> 5;
    int tap = k >> 6;
    int c   = k & 63;
    float v = 0.f;
    if (oc < 18)       v = ow[((size_t)(oc * CIN + c)) * 9 + tap];
    else if (oc < 27)  v = mw[((size_t)((oc - 18) * CIN + c)) * 9 + tap];
    wcomb[((size_t)(k >> 1) * 32 + oc) * 2 + (k & 1)] = v;
    if (k == 0)
        bias32[oc] = (oc < 18) ? ob[oc] : ((oc < 27) ? mb[oc - 18] : 0.f);
}

// ---------------------------------------------------------------------------
// Kernel 2: DCN weights -> pair-interleaved panel:
// wd[((K>>1)*64 + o)*2 + (K&1)] = dcn_w[o][c][tap], K = tap*64 + c.
// ---------------------------------------------------------------------------
__global__ __launch_bounds__(256) void pack_dcn_w(const float* __restrict__ dw,
                                                  float* __restrict__ wd) {
    int idx = blockIdx.x * 256 + threadIdx.x;
    if (idx >= KTOT * COUT) return;
    int o   = idx & 63;
    int k   = idx >> 6;
    int tap = k >> 6;
    int c   = k & 63;
    wd[((size_t)(k >> 1) * COUT + o) * 2 + (k & 1)] = dw[((size_t)(o * CIN + c)) * 9 + tap];
}

// ---------------------------------------------------------------------------
// Kernel 3: offset+mask 3x3 conv as implicit GEMM on V_WMMA_F32_16X16X4_F32.
// One wave per 16-pixel tile. M=16, N=32, K=576. Boundary handled with a 0/1
// float scale on clamped unconditional loads (no EXEC branching).
// ---------------------------------------------------------------------------
__global__ __launch_bounds__(256) void conv_offmask(const float* __restrict__ xt,
                                                    const float* __restrict__ wcomb,
                                                    const float* __restrict__ bias32,
                                                    float* __restrict__ offmask) {
    const int wave = blockIdx.x * 8 + (threadIdx.x >> 5);
    const int lane = threadIdx.x & 31;
    const int m    = lane & 15;   // A: row M | B/D: col N
    const int kh   = lane >> 4;   // K-pair select (A/B), M+8 half (D)
    const int p0   = wave * 16;
    const int b    = p0 >> 14;
    const int rem  = p0 & (HW - 1);
    const int h    = rem >> 7;
    const int w0   = rem & (Ws - 1);
    const int w    = w0 + m;

    v8f acc0 = {}, acc1 = {};
    const int n0 = m, n1 = m + 16;

    for (int tap = 0; tap < 9; ++tap) {
        const int y  = h + tap / 3 - 1;
        const int xx = w + tap % 3 - 1;
        const bool inb = (y >= 0) & (y < Hs) & (xx >= 0) & (xx < Ws);
        const float s = inb ? 1.f : 0.f;
        const int yc = min(max(y, 0), Hs - 1);
        const int xc = min(max(xx, 0), Ws - 1);
        const float* arow = xt + (((size_t)b * HW) + (size_t)yc * Ws + xc) * CIN + 2 * kh;
        // pair-base: kpair = tap*32 + c0/2 + kh
        const float* bbase = wcomb + ((size_t)(tap * 32 + kh) * 32) * 2;
#pragma unroll
        for (int c0 = 0; c0 < 64; c0 += 4) {
            v2f a = s * (*(const v2f*)(arow + c0));
            const float* bp = bbase + (size_t)(c0 >> 1) * 32 * 2;
            v2f b0 = *(const v2f*)(bp + 2 * n0);
            v2f b1 = *(const v2f*)(bp + 2 * n1);
            acc0 = __builtin_amdgcn_wmma_f32_16x16x4_f32(false, a, false, b0,
                                                         (short)0, acc0, false, false);
            acc1 = __builtin_amdgcn_wmma_f32_16x16x4_f32(false, a, false, b1,
                                                         (short)0, acc1, false, false);
        }
    }

    const float bia0 = bias32[n0];
    const float bia1 = bias32[n1];
#pragma unroll
    for (int r = 0; r < 8; ++r) {
        const int M = r + 8 * kh;
        const size_t p = (size_t)(p0 + M);
        offmask[p * 32 + n0] = acc0[r] + bia0;                    // n0 < 16: offsets
        float v1 = acc1[r] + bia1;
        if (n1 >= 18 && n1 < 27) v1 = 1.f / (1.f + __expf(-v1));  // masks -> sigmoid
        offmask[p * 32 + n1] = v1;
    }
}

// ---------------------------------------------------------------------------
// Per-tap bilinear descriptor: 4 clamped corner-row pointers + mask-folded
// weights (invalid corners get weight 0 == reference's g*valid semantics).
// ---------------------------------------------------------------------------
struct TapDesc {
    const float* r00; const float* r01; const float* r10; const float* r11;
    float w00, w01, w10, w11;
};

__device__ __forceinline__ TapDesc make_tap(const float* __restrict__ xt,
                                            size_t planebase, int h, int w,
                                            int ky0, int kx0,
                                            const float* __restrict__ om,
                                            int tap, int kh) {
    TapDesc t;
    const float dy  = om[2 * tap];
    const float dx  = om[2 * tap + 1];
    const float msk = om[18 + tap];
    const float py  = (float)(h + ky0) + dy;
    const float px  = (float)(w + kx0) + dx;
    const float fy0 = floorf(py), fx0 = floorf(px);
    const float ly = py - fy0, lx = px - fx0;
    const int y0 = (int)fy0, x0 = (int)fx0;
    const int y1 = y0 + 1,   x1 = x0 + 1;
    const bool vy0 = (y0 >= 0) & (y0 < Hs), vy1 = (y1 >= 0) & (y1 < Hs);
    const bool vx0 = (x0 >= 0) & (x0 < Ws), vx1 = (x1 >= 0) & (x1 < Ws);
    t.w00 = (vy0 & vx0) ? (1.f - ly) * (1.f - lx) * msk : 0.f;
    t.w01 = (vy0 & vx1) ? (1.f - ly) * lx         * msk : 0.f;
    t.w10 = (vy1 & vx0) ? ly         * (1.f - lx) * msk : 0.f;
    t.w11 = (vy1 & vx1) ? ly         * lx         * msk : 0.f;
    const int yc0 = min(max(y0, 0), Hs - 1), yc1 = min(max(y1, 0), Hs - 1);
    const int xc0 = min(max(x0, 0), Ws - 1), xc1 = min(max(x1, 0), Ws - 1);
    t.r00 = xt + (planebase + (size_t)yc0 * Ws + xc0) * CIN + 2 * kh;
    t.r01 = xt + (planebase + (size_t)yc0 * Ws + xc1) * CIN + 2 * kh;
    t.r10 = xt + (planebase + (size_t)yc1 * Ws + xc0) * CIN + 2 * kh;
    t.r11 = xt + (planebase + (size_t)yc1 * Ws + xc1) * CIN + 2 * kh;
    return t;
}

__device__ __forceinline__ v2f gather2(const TapDesc& t, int c0) {
    v2f g00 = *(const v2f*)(t.r00 + c0);
    v2f g01 = *(const v2f*)(t.r01 + c0);
    v2f g10 = *(const v2f*)(t.r10 + c0);
    v2f g11 = *(const v2f*)(t.r11 + c0);
    return t.w00 * g00 + t.w01 * g01 + t.w10 * g10 + t.w11 * g11;
}

// ---------------------------------------------------------------------------
// Kernel 4: deformable conv fused into implicit GEMM. Each wave computes a
// 32-pixel x 64-output tile (two 16x16 M tiles sharing every B fragment):
// per K-step, 2 A gathers + 4 shared B b64 loads feed 8 WMMAs.
// ---------------------------------------------------------------------------
__global__ __launch_bounds__(256) void deform_gemm(const float* __restrict__ xt,
                                                   const float* __restrict__ offmask,
                                                   const float* __restrict__ wd,
                                                   float* __restrict__ out) {
    const int wave = blockIdx.x * 8 + (threadIdx.x >> 5);
    const int lane = threadIdx.x & 31;
    const int m    = lane & 15;
    const int kh   = lane >> 4;
    const int p0   = wave * 32;           // 32 pixels per wave, same b,h
    const int b    = p0 >> 14;
    const int rem  = p0 & (HW - 1);
    const int h    = rem >> 7;
    const int w0   = rem & (Ws - 1);
    const int wA   = w0 + m;              // M tile 0 row
    const int wB   = w0 + 16 + m;         // M tile 1 row

    v8f accA[4] = {{}, {}, {}, {}};
    v8f accB[4] = {{}, {}, {}, {}};
    const float* omA = offmask + (size_t)(p0 + m) * 32;
    const float* omB = offmask + (size_t)(p0 + 16 + m) * 32;
    const size_t planebase = (size_t)b * HW;

    for (int tap = 0; tap < 9; ++tap) {
        const int ky0 = tap / 3 - 1;
        const int kx0 = tap % 3 - 1;
        TapDesc tA = make_tap(xt, planebase, h, wA, ky0, kx0, omA, tap, kh);
        TapDesc tB = make_tap(xt, planebase, h, wB, ky0, kx0, omB, tap, kh);
        const float* bbase = wd + ((size_t)(tap * 32 + kh) * COUT) * 2;
#pragma unroll
        for (int c0 = 0; c0 < 64; c0 += 4) {
            v2f aA = gather2(tA, c0);
            v2f aB = gather2(tB, c0);
            const float* bp = bbase + (size_t)(c0 >> 1) * COUT * 2;
#pragma unroll
            for (int nt = 0; nt < 4; ++nt) {
                v2f bf = *(const v2f*)(bp + 2 * (nt * 16 + m));
                accA[nt] = __builtin_amdgcn_wmma_f32_16x16x4_f32(false, aA, false, bf,
                                                                 (short)0, accA[nt], false, false);
                accB[nt] = __builtin_amdgcn_wmma_f32_16x16x4_f32(false, aB, false, bf,
                                                                 (short)0, accB[nt], false, false);
            }
        }
    }

#pragma unroll
    for (int nt = 0; nt < 4; ++nt) {
        const int o = nt * 16 + m;
        const size_t obase = ((size_t)(b * COUT + o)) * HW + (size_t)h * Ws;
#pragma unroll
        for (int r = 0; r < 8; ++r) {
            const int M = r + 8 * kh;
            out[obase + (w0 + M)]      = accA[nt][r];
            out[obase + (w0 + 16 + M)] = accB[nt][r];
        }
    }
}

// ---------------------------------------------------------------------------
extern "C" void kernel_launch(void* const* d_in, const int* in_sizes, int n_in,
                              void* d_out, int out_size, void* d_ws, size_t ws_size,
                              hipStream_t stream) {
    const float* x  = (const float*)d_in[0];   // (8,64,128,128)
    const float* ow = (const float*)d_in[1];   // (18,64,3,3)
    const float* ob = (const float*)d_in[2];   // (18,)
    const float* mw = (const float*)d_in[3];   // (9,64,3,3)
    const float* mb = (const float*)d_in[4];   // (9,)
    const float* dw = (const float*)d_in[5];   // (64,64,3,3)
    float* out = (float*)d_out;                // (8,64,128,128)

    // Workspace carve-out (~50.6 MB total)
    float* xt      = (float*)d_ws;                 // 8,388,608 floats (NHWC x)
    float* offmask = xt + (size_t)NPOS * CIN;      // 4,194,304 floats
    float* wcomb   = offmask + (size_t)NPOS * 32;  // 18,432 floats (pair-interleaved)
    float* bias32  = wcomb + KTOT * 32;            // 32 floats
    float* wd      = bias32 + 32;                  // 36,864 floats (pair-interleaved)

    nchw_to_nhwc<<<(NPOS * CIN) / 256, 256, 0, stream>>>(x, xt);
    pack_conv_w<<<(KTOT * 32 + 255) / 256, 256, 0, stream>>>(ow, ob, mw, mb, wcomb, bias32);
    pack_dcn_w<<<(KTOT * COUT + 255) / 256, 256, 0, stream>>>(dw, wd);

    conv_offmask<<<NPOS / 16 / 8, 256, 0, stream>>>(xt, wcomb, bias32, offmask);  // 1024 blocks
    deform_gemm<<<NPOS / 32 / 8, 256, 0, stream>>>(xt, offmask, wd, out);         // 512 blocks
}